// ChamferLoss_1580547969429
// MI455X (gfx1250) — compile-verified
//
#include <hip/hip_runtime.h>

typedef float v2f __attribute__((ext_vector_type(2)));
typedef float v8f __attribute__((ext_vector_type(8)));

#define BATCHES        8
#define NPTS           8192
#define ROWS_PER_BLOCK 128          // 8 waves * 16 rows
#define THREADS        256          // 8 wave32 waves
#define JTILES         (NPTS / 16)

#define INF_BITS 0x7F800000u

__global__ void init_ws(unsigned* colMin, float* out, int n) {
    int idx = blockIdx.x * blockDim.x + threadIdx.x;
    if (idx < n) colMin[idx] = INF_BITS;
    if (idx == 0) out[0] = 0.0f;
}

// Each block: one batch, 128 gts rows (16 per wave). Wave sweeps all 8192 preds
// in 16-wide tiles; each tile is ONE v_wmma_f32_16x16x4_f32.
//   A row i  = (-2gx, -2gy, -2gz, 1)        (2 VGPRs, loop-invariant)
//   B col j  = ( px,   py,   pz, |p|^2)     (2 VGPRs, ds_load_b64 per tile)
//   C[i][j]  = |g_i|^2                      (8 VGPRs, loop-invariant)
//   D        = |g|^2 + |p|^2 - 2 g.p  (squared distance tile)
//
// Column mins are clamped to >= 0 (true distances are non-negative; negatives
// are rounding noise), so their raw IEEE bits order as unsigned ints and we
// can use native integer min atomics (ds_min_u32 / global_atomic_min_u32).
__global__ __launch_bounds__(THREADS, 1)
void chamfer_tiles(const float* __restrict__ preds,
                   const float* __restrict__ gts,
                   float* __restrict__ out,
                   unsigned* __restrict__ colMinG) {
    __shared__ unsigned sColMin[NPTS];   // 32 KB: per-j min (bit keys)
    __shared__ float4   sPts[NPTS];      // 128 KB: preds batch as (x,y,z,|p|^2)

    const int tid   = threadIdx.x;
    const int lane  = tid & 31;
    const int m     = lane & 15;         // M (A) or N (B/C/D) position
    const int half  = lane >> 4;         // selects K-pair {0,1} vs {2,3}
    const int wave  = tid >> 5;
    const int batch = blockIdx.x / (NPTS / ROWS_PER_BLOCK);
    const int iBase = (blockIdx.x % (NPTS / ROWS_PER_BLOCK)) * ROWS_PER_BLOCK + wave * 16;

    // Preload preds batch -> LDS, init per-block column mins.
    const float* P = preds + (size_t)batch * NPTS * 3;
    for (int idx = tid; idx < NPTS; idx += THREADS) {
        float x = P[idx * 3 + 0], y = P[idx * 3 + 1], z = P[idx * 3 + 2];
        sPts[idx]    = make_float4(x, y, z, x * x + y * y + z * z);
        sColMin[idx] = INF_BITS;
    }
    __syncthreads();

    // A fragment (16x4 f32): row i = (-2gx, -2gy, -2gz, 1)
    const float* G = gts + ((size_t)batch * NPTS + (iBase + m)) * 3;
    float gx = G[0], gy = G[1], gz = G[2];
    float gsq = gx * gx + gy * gy + gz * gz;
    v2f a;
    a.x = half ? (-2.0f * gz) : (-2.0f * gx);   // K=2 : K=0
    a.y = half ? 1.0f         : (-2.0f * gy);   // K=3 : K=1

    // C fragment: C[v] = |g_{iBase + v + 8*half}|^2  (loop-invariant broadcast)
    v8f c;
#pragma unroll
    for (int v = 0; v < 8; ++v)
        c[v] = __shfl(gsq, v + 8 * half, 32);

    float rmin[8];
#pragma unroll
    for (int v = 0; v < 8; ++v) rmin[v] = 3.402823466e38f;

    // B is read as float2 halves of the stored float4: elem index 2*j + half.
    const float2* sB = (const float2*)sPts;
    int bi = m * 2 + half;                       // tile 0 index for this lane
    float2 bcur = sB[bi];

    for (int jt = 0; jt < JTILES; ++jt) {
        bi = (bi + 32) & (2 * NPTS - 1);         // wrap -> branch-free prefetch
        float2 bnxt = sB[bi];                    // ds_load_b64, hidden under compute

        v2f bf;
        bf.x = bcur.x;
        bf.y = bcur.y;
        v8f d = __builtin_amdgcn_wmma_f32_16x16x4_f32(
            false, a, false, bf, (short)0, c, false, false);

        // running row mins (rows v / v+8 depending on half)
#pragma unroll
        for (int v = 0; v < 8; ++v) rmin[v] = fminf(rmin[v], d[v]);

        // column min over this tile's rows: both halves atomically min the
        // same slot (2-way LDS address conflict; no shuffles / EXEC games).
        float cm = fminf(fminf(fminf(d[0], d[1]), fminf(d[2], d[3])),
                         fminf(fminf(d[4], d[5]), fminf(d[6], d[7])));
        cm = fmaxf(cm, 0.0f);                    // make bit-key ordering valid
        atomicMin(&sColMin[jt * 16 + m], __float_as_uint(cm));   // ds_min_u32

        bcur = bnxt;
    }

    // loss_2: per-row min over all j -> reduce each 16-lane half, sum, add.
#pragma unroll
    for (int v = 0; v < 8; ++v) {
        float r = rmin[v];
        r = fminf(r, __shfl_xor(r, 1, 32));
        r = fminf(r, __shfl_xor(r, 2, 32));
        r = fminf(r, __shfl_xor(r, 4, 32));
        r = fminf(r, __shfl_xor(r, 8, 32));
        rmin[v] = r;
    }
    float s = 0.0f;
#pragma unroll
    for (int v = 0; v < 8; ++v) s += rmin[v];
    if (m == 0) atomicAdd(out, s);               // lanes 0 (rows 0-7) and 16 (rows 8-15)

    __syncthreads();
    // Flush block-local column mins to global (one u32 atomicMin per j per block).
    unsigned* wsCol = colMinG + (size_t)batch * NPTS;
    for (int idx = tid; idx < NPTS; idx += THREADS)
        atomicMin(&wsCol[idx], sColMin[idx]);    // global_atomic_min_u32
}

// loss_1: sum the decoded global column mins into the scalar output.
__global__ void reduce_colmin(const unsigned* __restrict__ colMin,
                              float* __restrict__ out, int n) {
    int tid = blockIdx.x * blockDim.x + threadIdx.x;
    float s = 0.0f;
    for (int idx = tid; idx < n; idx += gridDim.x * blockDim.x)
        s += __uint_as_float(colMin[idx]);       // keys are non-negative floats
    s += __shfl_xor(s, 16, 32);
    s += __shfl_xor(s, 8, 32);
    s += __shfl_xor(s, 4, 32);
    s += __shfl_xor(s, 2, 32);
    s += __shfl_xor(s, 1, 32);
    if ((threadIdx.x & 31) == 0) atomicAdd(out, s);
}

extern "C" void kernel_launch(void* const* d_in, const int* in_sizes, int n_in,
                              void* d_out, int out_size, void* d_ws, size_t ws_size,
                              hipStream_t stream) {
    const float* preds = (const float*)d_in[0];   // [8, 8192, 3]
    const float* gts   = (const float*)d_in[1];   // [8, 8192, 3]
    float*    out    = (float*)d_out;             // scalar
    unsigned* colMin = (unsigned*)d_ws;           // 8*8192 u32 keys (256 KB)

    const int total = BATCHES * NPTS;
    init_ws<<<(total + 255) / 256, 256, 0, stream>>>(colMin, out, total);
    chamfer_tiles<<<BATCHES * (NPTS / ROWS_PER_BLOCK), THREADS, 0, stream>>>(
        preds, gts, out, colMin);
    reduce_colmin<<<64, 256, 0, stream>>>(colMin, out, total);
}